// ConvCapsules2d_1726576853902
// MI455X (gfx1250) — compile-verified
//
#include <hip/hip_runtime.h>
#include <hip/hip_bf16.h>

typedef __attribute__((ext_vector_type(2))) float v2f;
typedef __attribute__((ext_vector_type(8))) float v8f;

#define NB   8
#define BC   32
#define CC   32
#define PD   4
#define KS   3
#define STR  2
#define HH   14
#define FF   6
#define SPOS 324            // F*F*K*K
#define BLK  325            // padded LDS block stride (bank-conflict-free scatter)
#define ACT_SZ (NB*BC*SPOS) // 82944

__global__ __launch_bounds__(256) void capsconv_wmma_kernel(
    const float* __restrict__ acts,   // [N,B,14,14]
    const float* __restrict__ poses,  // [N,B,4,4,14,14]
    const float* __restrict__ W,      // [B,C,4,4,3,3]
    float* __restrict__ out)          // acts_out (82944) ++ V_ji (42467328)
{
    __shared__ float sP[PD*PD*HH*HH];    // 3136 : poses[n,b]
    __shared__ float sW[4*PD*PD*KS*KS];  //  576 : W[b, cg*4..cg*4+3]
    __shared__ float sO[4*16*BLK];       // 20800: padded output tile

    const int tid = threadIdx.x;
    const int blk = blockIdx.x;
    const int cg  = blk & 7;          // c-group: 4 output capsules
    const int b   = (blk >> 3) & 31;
    const int n   = blk >> 8;

    // ---- stage inputs to LDS (both slices are contiguous in memory) ----
    const float* gP = poses + (size_t)(n*BC + b) * (PD*PD*HH*HH);
    for (int t = tid; t < PD*PD*HH*HH; t += 256) sP[t] = gP[t];
    const float* gW = W + (size_t)(b*CC + cg*4) * (PD*PD*KS*KS);
    for (int t = tid; t < 4*PD*PD*KS*KS; t += 256) sW[t] = gW[t];
    __syncthreads();

    const int lane  = tid & 31;
    const int wave  = tid >> 5;
    const int half  = lane >> 4;   // K half for A/B fragments; M half for D
    const int l15   = lane & 15;
    // A-row decoding: M = fgsub*4 + i
    const int iA    = l15 & 3;
    const int fgsub = l15 >> 2;
    // B-col decoding: N = c_loc*4 + j
    const int c_loc = l15 >> 2;
    const int jB    = l15 & 3;
    const int m0    = half * 2;    // K rows owned by this half-wave

    const v8f c0 = {0.f,0.f,0.f,0.f,0.f,0.f,0.f,0.f};

    // 81 independent 16x16x4 tiles: jb -> (kl in 0..8, fg-quad in 0..8)
    for (int jb = wave; jb < 81; jb += 8) {
        const int kl  = jb / 9;
        const int fgq = jb - kl*9;
        const int kk  = kl / 3;
        const int ll  = kl - kk*3;

        // A fragment: a(M, K=m) = poses[n,b, i, m, f*2+k, g*2+l]
        const int fg = fgq*4 + fgsub;
        const int f  = fg / 6, g = fg - f*6;
        const int y  = f*STR + kk, x = g*STR + ll;
        v2f a;
        a.x = sP[((iA*4 + m0 + 0)*HH + y)*HH + x];
        a.y = sP[((iA*4 + m0 + 1)*HH + y)*HH + x];

        // B fragment: b(K=m, N) = W[b, cg*4+c_loc, m, j, k, l]
        v2f bf;
        bf.x = sW[((c_loc*4 + m0 + 0)*4 + jB)*9 + kl];
        bf.y = sW[((c_loc*4 + m0 + 1)*4 + jB)*9 + kl];

        v8f d = __builtin_amdgcn_wmma_f32_16x16x4_f32(
            false, a, false, bf, (short)0, c0, false, false);

        // scatter D into padded LDS tile: addr = (c_loc*16 + i*4 + j)*325 + f*54+g*9+kl
        // lane bank pattern (16*c_loc + 5*j) mod 64 -> conflict-free
        #pragma unroll
        for (int v = 0; v < 8; ++v) {
            const int M   = v + half*8;
            const int ii  = M & 3;
            const int fg2 = fgq*4 + (M >> 2);
            const int f2  = fg2 / 6, g2 = fg2 - f2*6;
            sO[(c_loc*16 + ii*4 + jB)*BLK + f2*54 + g2*9 + kl] = d[v];
        }
    }
    __syncthreads();

    // ---- coalesced 128-bit streaming store of the 20736-float tile ----
    float* gV = out + ACT_SZ + (size_t)(((n*BC + b)*CC + cg*4) * 16) * SPOS;
    for (int t = tid*4; t < 4*16*SPOS; t += 256*4) {
        const int q = t / SPOS;          // (c_loc,ij) block
        const int r = t - q*SPOS;        // multiple of 4, never crosses block end
        const float* s = &sO[q*BLK + r];
        float4 val = make_float4(s[0], s[1], s[2], s[3]);
        *(float4*)(gV + t) = val;
    }

    // ---- acts_out: tiny gather, emitted once per (n,b) by cg==0 blocks ----
    if (cg == 0 && tid < SPOS) {
        const int f = tid / 54;
        int r = tid - f*54;
        const int g = r / 9;  r -= g*9;
        const int kk = r / 3, ll = r - kk*3;
        out[(n*BC + b)*SPOS + tid] =
            acts[((n*BC + b)*HH + (f*STR + kk))*HH + (g*STR + ll)];
    }
}

extern "C" void kernel_launch(void* const* d_in, const int* in_sizes, int n_in,
                              void* d_out, int out_size, void* d_ws, size_t ws_size,
                              hipStream_t stream) {
    const float* acts  = (const float*)d_in[0]; // [8,32,14,14]
    const float* poses = (const float*)d_in[1]; // [8,32,4,4,14,14]
    const float* W     = (const float*)d_in[2]; // [32,32,4,4,3,3]
    float* out = (float*)d_out;                 // 82944 + 42467328 floats

    dim3 grid(NB * BC * 8);   // (n, b, c-group)
    dim3 block(256);          // 8 wave32
    capsconv_wmma_kernel<<<grid, block, 0, stream>>>(acts, poses, W, out);
}